// GraphSearcherMerged_18219251269838
// MI455X (gfx1250) — compile-verified
//
#include <hip/hip_runtime.h>

typedef __attribute__((ext_vector_type(2))) float v2f;
typedef __attribute__((ext_vector_type(8))) float v8f;

#define NROWS  1000000
#define NSITES 250000
#define DIN    128
#define DFEAT  128
#define DOUT   64

// ---------------------------------------------------------------------------
// Tiny prep: Wd = W1[:128,:] - W1[128:,:]   (8192 floats, L2-resident forever)
// Lets us rewrite  out = feat@W1a + (ss[seg]-feat)@W1b + b1
//                      = feat@Wd + ss[seg]@W1b + b1
// so `features` NEVER has to be materialized in HBM (saves ~1.5 GB traffic).
// ---------------------------------------------------------------------------
__global__ __launch_bounds__(256) void wd_kernel(const float* __restrict__ W1,
                                                 float* __restrict__ Wd)
{
    int i = blockIdx.x * 256 + threadIdx.x;          // DFEAT*DOUT = 8192
    if (i < DFEAT * DOUT)
        Wd[i] = W1[i] - W1[DFEAT * DOUT + i];
}

// ---------------------------------------------------------------------------
// Fused kernel: per 16-row tile (256 threads = 8 waves)
//   Phase 1 (all 8 waves): feat = relu(X_tile @ W0 + b0) via f32 WMMA,
//            accumulator written straight to LDS (never to HBM).
//   Phase 2 (wave-split, concurrent):
//     waves 0-3: sorted-segment boundary-flush atomics into site_sum
//                (one thread per feature column, ~1 atomic per segment)
//     waves 4-7: out_tile = feat_tile @ Wd + b1 via f32 WMMA
//                (atomics latency hidden under the WMMA loop)
// ---------------------------------------------------------------------------
__global__ __launch_bounds__(256) void fused0_kernel(
    const float* __restrict__ X, const float* __restrict__ W0,
    const float* __restrict__ b0, const float* __restrict__ Wd,
    const float* __restrict__ b1, const int* __restrict__ seg,
    float* __restrict__ site_sum, float* __restrict__ out)
{
    __shared__ float Xs[16][DIN + 4];    // +4 pad: conflict-free strided A reads
    __shared__ float Fs[16][DFEAT + 4];  // feat tile lives only in LDS
    __shared__ int   segs[16];

    const int r0  = blockIdx.x * 16;
    const int tid = threadIdx.x;
    if (tid < 16) segs[tid] = seg[r0 + tid];

    // Coalesced cooperative load of the 16x128 fp32 X tile (8 KB)
    #pragma unroll
    for (int i = 0; i < 8; ++i) {
        int e = tid + i * 256;
        int row = e >> 7, col = e & 127;
        Xs[row][col] = X[(size_t)(r0 + row) * DIN + col];
    }
    __syncthreads();

    const int lane = tid & 31;
    const int wave = tid >> 5;
    const int n0   = wave * 16;                // feat column tile (8 waves x 16)
    const int ml   = lane & 15;                // A: M this lane holds
    const int nl   = lane & 15;                // B/C/D: N this lane holds
    const int kb   = (lane >> 4) << 1;         // A/B: K sub-offset (0 or 2)
    const int mofs = (lane >> 4) << 3;         // lanes 16-31 hold M = i + 8

    // ---- Phase 1: feat tile ------------------------------------------------
    v8f c = {0.f, 0.f, 0.f, 0.f, 0.f, 0.f, 0.f, 0.f};
    #pragma unroll 4
    for (int k0 = 0; k0 < DIN; k0 += 4) {
        v2f a, b;
        a.x = Xs[ml][k0 + kb];
        a.y = Xs[ml][k0 + kb + 1];
        b.x = W0[(size_t)(k0 + kb)     * DFEAT + n0 + nl];   // W0: 64 KB, L2-hot
        b.y = W0[(size_t)(k0 + kb + 1) * DFEAT + n0 + nl];
        c = __builtin_amdgcn_wmma_f32_16x16x4_f32(false, a, false, b,
                                                  (short)0, c, false, false);
    }
    const float bias0 = b0[n0 + nl];
    #pragma unroll
    for (int i = 0; i < 8; ++i) {
        float v = c[i] + bias0;
        v = v > 0.f ? v : 0.f;                 // fused ReLU
        Fs[i + mofs][n0 + nl] = v;             // stays in LDS
    }
    __syncthreads();

    // ---- Phase 2: wave-split -----------------------------------------------
    if (wave < 4) {
        // Segment sum: sorted ids -> register accumulate, atomic only at
        // boundaries. Cross-block splits are safe (every flush is atomic).
        const int f   = tid;                   // feature column 0..127
        int   cur = segs[0];
        float run = Fs[0][f];
        for (int r = 1; r < 16; ++r) {
            int s = segs[r];
            if (s != cur) {
                unsafeAtomicAdd(&site_sum[(size_t)cur * DFEAT + f], run);
                run = 0.f;
                cur = s;
            }
            run += Fs[r][f];
        }
        unsafeAtomicAdd(&site_sum[(size_t)cur * DFEAT + f], run);
    } else {
        // Partial output: feat_tile @ Wd + b1   (4 col tiles of 16 = DOUT)
        const int o0 = (wave - 4) * 16;
        v8f d = {0.f, 0.f, 0.f, 0.f, 0.f, 0.f, 0.f, 0.f};
        #pragma unroll 4
        for (int k0 = 0; k0 < DFEAT; k0 += 4) {
            v2f a, b;
            a.x = Fs[ml][k0 + kb];
            a.y = Fs[ml][k0 + kb + 1];
            b.x = Wd[(size_t)(k0 + kb)     * DOUT + o0 + nl];  // Wd: 32 KB, L2-hot
            b.y = Wd[(size_t)(k0 + kb + 1) * DOUT + o0 + nl];
            d = __builtin_amdgcn_wmma_f32_16x16x4_f32(false, a, false, b,
                                                      (short)0, d, false, false);
        }
        const float bias1 = b1[o0 + nl];
        #pragma unroll
        for (int i = 0; i < 8; ++i)
            out[(size_t)(r0 + i + mofs) * DOUT + o0 + nl] = d[i] + bias1;
    }
}

// ---------------------------------------------------------------------------
// Second pass: out += site_sum[seg] @ W1b      (K=128, N=64)
// site_sum (128 MB) fits in the 192 MB global L2, so the per-row gather is
// L2-resident after first touch; segment locality (avg 4 repeats) helps too.
// ---------------------------------------------------------------------------
__global__ __launch_bounds__(128) void gemm_ss_kernel(
    const float* __restrict__ site_sum, const int* __restrict__ seg,
    const float* __restrict__ W1, float* __restrict__ out)
{
    __shared__ float Ss[16][DFEAT + 4];
    __shared__ int   segs[16];
    const int r0  = blockIdx.x * 16;
    const int tid = threadIdx.x;
    if (tid < 16) segs[tid] = seg[r0 + tid];
    __syncthreads();

    #pragma unroll
    for (int i = 0; i < 16; ++i) {
        int e = tid + i * 128;
        int row = e >> 7, col = e & 127;
        Ss[row][col] = site_sum[(size_t)segs[row] * DFEAT + col];
    }
    __syncthreads();

    const int lane = tid & 31;
    const int wave = tid >> 5;
    const int n0   = wave * 16;
    const int ml   = lane & 15;
    const int nl   = lane & 15;
    const int kb   = (lane >> 4) << 1;
    const int mofs = (lane >> 4) << 3;

    v8f d = {0.f, 0.f, 0.f, 0.f, 0.f, 0.f, 0.f, 0.f};
    #pragma unroll 4
    for (int k0 = 0; k0 < DFEAT; k0 += 4) {
        v2f a, b;
        a.x = Ss[ml][k0 + kb];
        a.y = Ss[ml][k0 + kb + 1];
        b.x = W1[(size_t)(DFEAT + k0 + kb)     * DOUT + n0 + nl];  // W1b half
        b.y = W1[(size_t)(DFEAT + k0 + kb + 1) * DOUT + n0 + nl];
        d = __builtin_amdgcn_wmma_f32_16x16x4_f32(false, a, false, b,
                                                  (short)0, d, false, false);
    }

    #pragma unroll
    for (int i = 0; i < 8; ++i) {
        size_t idx = (size_t)(r0 + i + mofs) * DOUT + n0 + nl;
        out[idx] += d[i];                      // tile exclusively owns these rows
    }
}

// ---------------------------------------------------------------------------
// Launch. Workspace: Wd (32 KB) | site_sum (128 MB)  -- feat never hits HBM.
// ---------------------------------------------------------------------------
extern "C" void kernel_launch(void* const* d_in, const int* in_sizes, int n_in,
                              void* d_out, int out_size, void* d_ws, size_t ws_size,
                              hipStream_t stream) {
    const float* X   = (const float*)d_in[0];
    const int*   seg = (const int*)  d_in[1];
    const float* W0  = (const float*)d_in[2];
    const float* b0  = (const float*)d_in[3];
    const float* W1  = (const float*)d_in[4];
    const float* b1  = (const float*)d_in[5];
    float*       out = (float*)d_out;

    float* Wd       = (float*)d_ws;
    float* site_sum = Wd + DFEAT * DOUT;       // 32 KB in, stays aligned

    hipMemsetAsync(site_sum, 0, (size_t)NSITES * DFEAT * sizeof(float), stream);
    wd_kernel   <<<(DFEAT * DOUT + 255) / 256, 256, 0, stream>>>(W1, Wd);
    fused0_kernel<<<NROWS / 16, 256, 0, stream>>>(X, W0, b0, Wd, b1, seg,
                                                  site_sum, out);
    gemm_ss_kernel<<<NROWS / 16, 128, 0, stream>>>(site_sum, seg, W1, out);
}